// CapsuleLayerTSV_22093311770686
// MI455X (gfx1250) — compile-verified
//
#include <hip/hip_runtime.h>
#include <math.h>

typedef __attribute__((ext_vector_type(2))) float v2f;
typedef __attribute__((ext_vector_type(8))) float v8f;

namespace {
constexpr int B_DIM = 256;
constexpr int R_DIM = 16;
constexpr int L_DIM = 300;
constexpr int K_DIM = 900;   // 3*L
constexpr int C_DIM = 3;
constexpr int F_DIM = 768;
constexpr int NTILE = 5;     // ceil(300/64) N-tiles of width 64 per wave
constexpr int MTILE = 8;     // 256/32  -> 32 M-rows per wave
constexpr int TILES_TOTAL = C_DIM * R_DIM * MTILE * NTILE; // 1920 waves
}

// -------- Stage 1: priors[c][b][r][j] = sum_i x[b][r][i] * W[c][r][i][j]
// 48 GEMMs (c,r): 256x900 @ 900x300, fp32 via V_WMMA_F32_16X16X4_F32.
// One wave -> 32(M) x 64(N) tile: 2 A fragments x 4 B fragments = 8 f32
// accumulators; B fragments are shared across both M sub-tiles so each
// 4-K step does 10 VMEM loads for 8 WMMAs (16384 FLOP).
__global__ __launch_bounds__(256) void priors_wmma_kernel(
    const float* __restrict__ x, const float* __restrict__ w,
    float* __restrict__ priors)
{
  const int lane = threadIdx.x & 31;
  const int wave = threadIdx.x >> 5;
  const int tile = blockIdx.x * 8 + wave;
  if (tile >= TILES_TOTAL) return;

  const int nt = tile % NTILE;
  const int mt = (tile / NTILE) % MTILE;
  const int r  = (tile / (NTILE * MTILE)) % R_DIM;
  const int c  =  tile / (NTILE * MTILE * R_DIM);

  const int lo16 = lane & 15;
  const int hi   = lane >> 4;      // 0 for lanes 0-15, 1 for lanes 16-31
  const int koff = hi * 2;         // A/B: lanes 0-15 hold K=0,1; 16-31 hold K=2,3

  const int row0 = mt * 32 + lo16;         // M rows for the two A fragments
  const float* xrow0 = x + (size_t)(row0 * R_DIM + r) * K_DIM;
  const float* xrow1 = xrow0 + (size_t)16 * R_DIM * K_DIM;
  const float* wmat  = w + (size_t)(c * R_DIM + r) * K_DIM * L_DIM;

  // Branchless N edge handling (keep EXEC all-1s around WMMA):
  int   jc[4];
  float jm[4];
#pragma unroll
  for (int s4 = 0; s4 < 4; ++s4) {
    int j = nt * 64 + s4 * 16 + lo16;
    jc[s4] = (j < L_DIM) ? j : (L_DIM - 1);
    jm[s4] = (j < L_DIM) ? 1.0f : 0.0f;
  }

  v8f acc[2][4] = {{v8f{}, v8f{}, v8f{}, v8f{}},
                   {v8f{}, v8f{}, v8f{}, v8f{}}};
  for (int k = 0; k < K_DIM; k += 4) {
    v2f a0, a1;
    a0.x = xrow0[k + koff + 0];
    a0.y = xrow0[k + koff + 1];
    a1.x = xrow1[k + koff + 0];
    a1.y = xrow1[k + koff + 1];
    const float* wk = wmat + (size_t)(k + koff) * L_DIM;
#pragma unroll
    for (int s4 = 0; s4 < 4; ++s4) {
      v2f bfrag;
      bfrag.x = wk[jc[s4]] * jm[s4];
      bfrag.y = wk[L_DIM + jc[s4]] * jm[s4];
      acc[0][s4] = __builtin_amdgcn_wmma_f32_16x16x4_f32(
          false, a0, false, bfrag, (short)0, acc[0][s4], false, false);
      acc[1][s4] = __builtin_amdgcn_wmma_f32_16x16x4_f32(
          false, a1, false, bfrag, (short)0, acc[1][s4], false, false);
    }
  }

  // D layout: VGPR v -> M = v (lanes 0-15) / v+8 (lanes 16-31), N = lane&15
#pragma unroll
  for (int mi = 0; mi < 2; ++mi) {
#pragma unroll
    for (int s4 = 0; s4 < 4; ++s4) {
      const int j = nt * 64 + s4 * 16 + lo16;
      if (j < L_DIM) {
#pragma unroll
        for (int v = 0; v < 8; ++v) {
          const int rowOut = mt * 32 + mi * 16 + v + hi * 8;
          priors[(((size_t)c * B_DIM + rowOut) * R_DIM + r) * L_DIM + j] =
              acc[mi][s4][v];
        }
      }
    }
  }
}

// -------- Stage 2: dynamic routing, one workgroup per (c, b).
__global__ __launch_bounds__(256) void routing_kernel(
    const float* __restrict__ priors, float* __restrict__ vote_out)
{
  __shared__ float P[R_DIM * L_DIM];   // 19.2 KB
  __shared__ float vote[L_DIM];
  __shared__ float outs[L_DIM];
  __shared__ float logits[R_DIM];
  __shared__ float probs[R_DIM];
  __shared__ float red[8];
  __shared__ float snv;

  const int tid  = threadIdx.x;
  const int lane = tid & 31;
  const int wave = tid >> 5;
  const int c = blockIdx.x / B_DIM;
  const int b = blockIdx.x % B_DIM;

  const float* pin = priors + ((size_t)c * B_DIM + b) * R_DIM * L_DIM;
  for (int i = tid; i < R_DIM * L_DIM; i += 256) P[i] = pin[i];
  if (tid < R_DIM) logits[tid] = 0.0f;
  __syncthreads();

  for (int it = 0; it < 3; ++it) {
    // softmax over r (16 values; serial on thread 0 is negligible)
    if (tid == 0) {
      float m = logits[0];
      for (int rr = 1; rr < R_DIM; ++rr) m = fmaxf(m, logits[rr]);
      float ssum = 0.0f;
      for (int rr = 0; rr < R_DIM; ++rr) {
        float e = __expf(logits[rr] - m);
        probs[rr] = e;
        ssum += e;
      }
      float inv = 1.0f / ssum;
      for (int rr = 0; rr < R_DIM; ++rr) probs[rr] *= inv;
    }
    __syncthreads();

    // vote[j] = sum_r probs[r] * P[r][j]
    for (int j = tid; j < L_DIM; j += 256) {
      float v = 0.0f;
#pragma unroll
      for (int rr = 0; rr < R_DIM; ++rr) v += probs[rr] * P[rr * L_DIM + j];
      vote[j] = v;
    }
    __syncthreads();

    if (it == 2) break;  // final vote_outputs is saved before squash

    // sn = sum_j vote[j]^2  (wave32 shuffle reduction + cross-wave in LDS)
    float part = 0.0f;
    for (int j = tid; j < L_DIM; j += 256) part += vote[j] * vote[j];
#pragma unroll
    for (int off = 16; off > 0; off >>= 1) part += __shfl_xor(part, off, 32);
    if (lane == 0) red[wave] = part;
    __syncthreads();
    if (tid == 0) {
      float sn = 0.0f;
      for (int w8 = 0; w8 < 8; ++w8) sn += red[w8];
      snv = sn;
    }
    __syncthreads();

    const float sn = snv;
    const float scale = sqrtf(sn) / (1.0f + sn);  // == sn/(1+sn)/sqrt(sn)
    for (int j = tid; j < L_DIM; j += 256) outs[j] = vote[j] * scale;
    __syncthreads();

    // logits[r] += dot(P[r], outs); wave w owns r = 2w, 2w+1
#pragma unroll
    for (int rr2 = 0; rr2 < 2; ++rr2) {
      const int rr = wave * 2 + rr2;
      float p = 0.0f;
      for (int j = lane; j < L_DIM; j += 32) p += P[rr * L_DIM + j] * outs[j];
#pragma unroll
      for (int off = 16; off > 0; off >>= 1) p += __shfl_xor(p, off, 32);
      if (lane == 0) logits[rr] += p;
    }
    __syncthreads();
  }

  float* vo = vote_out + ((size_t)c * B_DIM + b) * L_DIM;
  for (int j = tid; j < L_DIM; j += 256) vo[j] = vote[j];
}

// -------- Stage 3: h = reshape(vote,(B,L,3)) @ larger_W.T + b, * sigmoid(s*elarger[t])
// 236 MB streaming output -> nontemporal stores (output > 192 MB L2).
__global__ __launch_bounds__(256) void final_expand_kernel(
    const float* __restrict__ vote, const float* __restrict__ lW,
    const float* __restrict__ lb, const float* __restrict__ elarger,
    const int* __restrict__ tptr, const int* __restrict__ sptr,
    float* __restrict__ out)
{
  const int o = blockIdx.x * 256 + threadIdx.x;  // 58,982,400 total, fits int
  const int f  = o % F_DIM;
  const int bl = o / F_DIM;              // b*300 + l
  const int b  = bl / L_DIM;
  const int l  = bl % L_DIM;
  // flat reshape (3,256,1,1,300) -> (256,300,3): h[b,l,cc] = vote_flat[b*900+l*3+cc]
  const int nbase = b * 900 + l * 3;
  const float h0 = vote[nbase + 0];
  const float h1 = vote[nbase + 1];
  const float h2 = vote[nbase + 2];
  const float acc =
      fmaf(h0, lW[f * 3 + 0], fmaf(h1, lW[f * 3 + 1], fmaf(h2, lW[f * 3 + 2], lb[f])));
  const int   t = tptr[0];
  const float s = (float)sptr[0];
  const float g = 1.0f / (1.0f + __expf(-s * elarger[t * F_DIM + f]));
  __builtin_nontemporal_store(acc * g, &out[o]);
}

extern "C" void kernel_launch(void* const* d_in, const int* in_sizes, int n_in,
                              void* d_out, int out_size, void* d_ws, size_t ws_size,
                              hipStream_t stream) {
  const float* x  = (const float*)d_in[0];   // (256,16,900)
  const float* rw = (const float*)d_in[1];   // (3,16,900,300)
  const float* el = (const float*)d_in[2];   // (16,768)
  const float* lW = (const float*)d_in[3];   // (768,3)
  const float* lb = (const float*)d_in[4];   // (768,)
  const int*   t  = (const int*)d_in[5];
  const int*   s  = (const int*)d_in[6];
  float* out = (float*)d_out;                // (256,300,768)

  float* priors = (float*)d_ws;                                   // 3*256*16*300 f32
  float* vote   = priors + (size_t)C_DIM * B_DIM * R_DIM * L_DIM; // 3*256*300 f32

  // 1920 tiles, 8 waves per 256-thread block
  priors_wmma_kernel<<<TILES_TOTAL / 8, 256, 0, stream>>>(x, rw, priors);
  routing_kernel<<<C_DIM * B_DIM, 256, 0, stream>>>(priors, vote);
  final_expand_kernel<<<(B_DIM * L_DIM * F_DIM) / 256, 256, 0, stream>>>(
      vote, lW, lb, el, t, s, out);
}